// DeltaNet_22488448762128
// MI455X (gfx1250) — compile-verified
//
#include <hip/hip_runtime.h>
#include <math.h>
#include <stdint.h>

// ---------------------------------------------------------------------------
// DeltaNet on MI455X (gfx1250, wave32)
//   B=2, L=2048, HID=2048, H=16, D=128
// Pipeline:
//   1) q = x@Wq, k = x@Wk, v = x@Wv        (WMMA f32 16x16x4, async-LDS GEMM)
//   2) beta = clip(sigmoid(x@Wg + bg))     (skinny GEMM, VALU)
//   3) RoPE + (elu+1) on q,k in place
//   4) sequential delta-rule scan, S kept in VGPRs (128 regs/thread)
//   5) out = y@Wo + bo                     (same WMMA GEMM, with bias)
//
// GEMM data path: GLOBAL_LOAD_ASYNC_TO_LDS_B128 (ASYNCcnt-tracked DMA,
// no VGPR round trip) double-buffered against the v_wmma stream;
// s_wait_asynccnt 0 + workgroup barrier at each K-tile boundary.
// ---------------------------------------------------------------------------

#define B_   2
#define L_   2048
#define HID_ 2048
#define H_   16
#define D_   128
#define EPS_ 1e-6f
#define BETA_MIN_ 0.8f
#define BETA_MAX_ 0.9995f

typedef __attribute__((ext_vector_type(2))) float v2f;
typedef __attribute__((ext_vector_type(8))) float v8f;

// ---- CDNA5 async global->LDS copy (16B) + ASYNCcnt wait (inline asm) ------
__device__ __forceinline__ void async_copy_b128(uint32_t lds_byte_addr,
                                                const void* gaddr) {
  asm volatile("global_load_async_to_lds_b128 %0, %1, off"
               :
               : "v"(lds_byte_addr), "v"((unsigned long long)(uintptr_t)gaddr)
               : "memory");
}
__device__ __forceinline__ void wait_async0() {
  asm volatile("s_wait_asynccnt 0" ::: "memory");
}

// =====================  WMMA fp32 GEMM: C = A*B (+bias)  ====================
// A: MxK row-major, B: KxN row-major, C: MxN row-major.
// Block = 256 threads (8 waves). Block tile 128x64, K-tile 32, double-buffered.
// Wave w computes rows [16w,16w+16) x all 64 cols -> 4 v8f accumulators.
#define TM 128
#define TN 64
#define TK 32
#define ASTRIDE 36   // 144B rows: 16B-aligned for B128, bank-conflict-free reads
#define BSTRIDE 72   // 288B rows: 16B-aligned, +2-row lane-half lands +16 banks

__global__ __launch_bounds__(256)
void gemm_f32_wmma(const float* __restrict__ A, const float* __restrict__ Bm,
                   const float* __restrict__ bias, float* __restrict__ C,
                   int M, int N, int K) {
  __shared__ float As[2][TM][ASTRIDE];   // 2*128*36*4 = 36864 B
  __shared__ float Bs[2][TK][BSTRIDE];   // 2*32*72*4  = 18432 B  (total 55296 B)

  const int m0 = blockIdx.y * TM;
  const int n0 = blockIdx.x * TN;
  const int tid = threadIdx.x;
  const int wid = tid >> 5;
  const int lane = tid & 31;
  const int lh = lane & 15;
  const int koff = (lane < 16) ? 0 : 2;   // ISA A/B layout: lane-halves hold K 0,1 / 2,3
  const int arow = wid * 16 + lh;

  v8f acc[4];
#pragma unroll
  for (int j = 0; j < 4; ++j) acc[j] = (v8f)0.0f;

  // ---- async-issue one K-tile (A: 4x B128/thread, B: 2x B128/thread) ----
  auto issue_tile = [&](int buf, int k0) {
#pragma unroll
    for (int it = 0; it < 4; ++it) {              // A tile: 128x32 = 1024 chunks
      int c = tid + it * 256;
      int row = c >> 3;
      int c4 = (c & 7) << 2;
      async_copy_b128((uint32_t)(uintptr_t)&As[buf][row][c4],
                      A + (size_t)(m0 + row) * K + k0 + c4);
    }
#pragma unroll
    for (int it = 0; it < 2; ++it) {              // B tile: 32x64 = 512 chunks
      int c = tid + it * 256;
      int row = c >> 4;
      int c4 = (c & 15) << 2;
      async_copy_b128((uint32_t)(uintptr_t)&Bs[buf][row][c4],
                      Bm + (size_t)(k0 + row) * N + n0 + c4);
    }
  };

  const int nk = K / TK;
  issue_tile(0, 0);

  for (int ik = 0; ik < nk; ++ik) {
    const int cur = ik & 1;
    wait_async0();        // this wave's DMA for tile `cur` has landed in LDS
    __syncthreads();      // ...and every other wave's too; readers of cur^1 done

    if (ik + 1 < nk) issue_tile(cur ^ 1, (ik + 1) * TK);

#pragma unroll
    for (int kk = 0; kk < TK; kk += 4) {
      v2f a;
      a[0] = As[cur][arow][kk + koff];
      a[1] = As[cur][arow][kk + koff + 1];
#pragma unroll
      for (int j = 0; j < 4; ++j) {
        v2f b;
        b[0] = Bs[cur][kk + koff][j * 16 + lh];
        b[1] = Bs[cur][kk + koff + 1][j * 16 + lh];
        acc[j] = __builtin_amdgcn_wmma_f32_16x16x4_f32(
            /*neg_a=*/false, a, /*neg_b=*/false, b,
            /*c_mod=*/(short)0, acc[j], /*reuse_a=*/false, /*reuse_b=*/false);
      }
    }
  }

  // ---- epilogue: C/D layout -> VGPR i: lanes0-15 M=i, lanes16-31 M=i+8 ----
  const int rbase = m0 + wid * 16 + ((lane < 16) ? 0 : 8);
#pragma unroll
  for (int j = 0; j < 4; ++j) {
    const int col = n0 + j * 16 + lh;
    const float bv = bias ? bias[col] : 0.0f;
#pragma unroll
    for (int i = 0; i < 8; ++i) {
      C[(size_t)(rbase + i) * N + col] = acc[j][i] + bv;
    }
  }
}

// ==================  beta = clip(sigmoid(x@Wg + bg))  =======================
// One block per (b,l). 256 threads = 16 heads x 16 partial lanes.
// Output layout: beta[(b*H + h)*L + l]  (scan-friendly).
__global__ __launch_bounds__(256)
void beta_kernel(const float* __restrict__ x, const float* __restrict__ Wg,
                 const float* __restrict__ bg, float* __restrict__ beta) {
  const int bl = blockIdx.x;                 // b*L + l
  const int h = threadIdx.x & 15;
  const int s = threadIdx.x >> 4;            // 0..15
  const float* xr = x + (size_t)bl * HID_;

  float acc = 0.0f;
  for (int c = s; c < HID_; c += 16)
    acc = fmaf(xr[c], Wg[(size_t)c * H_ + h], acc);

  __shared__ float part[16][16];
  part[s][h] = acc;
  __syncthreads();
  if (s == 0) {
    float v = bg[h];
#pragma unroll
    for (int ss = 0; ss < 16; ++ss) v += part[ss][h];
    v = 1.0f / (1.0f + expf(-v));
    v = fminf(fmaxf(v, BETA_MIN_), BETA_MAX_);
    const int b = bl / L_, l = bl % L_;
    beta[((size_t)b * H_ + h) * L_ + l] = v;
  }
}

// ==================  RoPE + phi = elu(.)+1 on q and k  ======================
// grid = B*L*H blocks, 64 threads; thread d handles the (d, d+64) pair.
__global__ __launch_bounds__(64)
void rope_phi_kernel(float* __restrict__ q, float* __restrict__ k) {
  const int idx = blockIdx.x;                // (b*L + t)*H + h
  const int t = (idx / H_) % L_;
  const int d = threadIdx.x;                 // 0..63
  const size_t base = (size_t)(idx / H_) * HID_ + (size_t)(idx % H_) * D_;

  // inv_freq = ROPE_BASE^(-d/64)
  const float inv = expf(-(float)d * (9.210340371976184f / 64.0f)); // ln(10000)
  const float ang = (float)t * inv;
  const float c = cosf(ang), s = sinf(ang);

  {
    float xe = q[base + d], xo = q[base + d + 64];
    float re = xe * c - xo * s;
    float ro = xe * s + xo * c;
    q[base + d]      = (re > 0.0f) ? (re + 1.0f) : expf(re);  // elu+1
    q[base + d + 64] = (ro > 0.0f) ? (ro + 1.0f) : expf(ro);
  }
  {
    float xe = k[base + d], xo = k[base + d + 64];
    float re = xe * c - xo * s;
    float ro = xe * s + xo * c;
    k[base + d]      = (re > 0.0f) ? (re + 1.0f) : expf(re);
    k[base + d + 64] = (ro > 0.0f) ? (ro + 1.0f) : expf(ro);
  }
}

// =====================  sequential delta-rule scan  =========================
// grid = B*H blocks, 128 threads. Thread e owns column e of S (128 VGPRs).
// y is written in place over v (safe: each (t,e) element of v is read only by
// the thread that overwrites it, before the overwrite).
__global__ __launch_bounds__(128)
void scan_kernel(const float* __restrict__ phi_q, const float* __restrict__ phi_k,
                 float* __restrict__ v_y, const float* __restrict__ beta) {
  const int bh = blockIdx.x;
  const int b = bh / H_, h = bh % H_;
  const int e = threadIdx.x;                 // 0..127 (column of S)

  __shared__ float ks[D_], qs[D_], red[4];

  float S[D_];
#pragma unroll
  for (int d = 0; d < D_; ++d) S[d] = 0.0f;
  float z = 0.0f;

  const float* bp = beta + (size_t)bh * L_;
  const size_t rowbase = (size_t)b * L_ * HID_ + (size_t)h * D_;

  for (int t = 0; t < L_; ++t) {
    const size_t off = rowbase + (size_t)t * HID_;
    const float bt = bp[t];
    ks[e] = phi_k[off + e];
    qs[e] = phi_q[off + e];
    const float ve = v_y[off + e];
    __syncthreads();

    const float ke = ks[e];
    const float qe = qs[e];
    z = z * bt + ke;
    float p = qe * z;
#pragma unroll
    for (int o = 16; o > 0; o >>= 1) p += __shfl_xor(p, o, 32);
    if ((e & 31) == 0) red[e >> 5] = p;
    __syncthreads();
    const float denom = red[0] + red[1] + red[2] + red[3] + EPS_;

    float numer = 0.0f;
#pragma unroll
    for (int d = 0; d < D_; ++d) {
      S[d] = S[d] * bt + ks[d] * ve;     // S = S*b + k (outer) v
      numer = fmaf(qs[d], S[d], numer);  // q . S[:,e]
    }
    v_y[off + e] = numer / denom;
    __syncthreads();                     // ks/qs reuse guard for next step
  }
}

// ============================  launcher  ====================================
extern "C" void kernel_launch(void* const* d_in, const int* in_sizes, int n_in,
                              void* d_out, int out_size, void* d_ws, size_t ws_size,
                              hipStream_t stream) {
  const float* x  = (const float*)d_in[0];
  const float* Wq = (const float*)d_in[1];
  const float* Wk = (const float*)d_in[2];
  const float* Wv = (const float*)d_in[3];
  const float* Wg = (const float*)d_in[4];
  const float* bg = (const float*)d_in[5];
  const float* Wo = (const float*)d_in[6];
  const float* bo = (const float*)d_in[7];
  float* out = (float*)d_out;

  const size_t nact = (size_t)B_ * L_ * HID_;        // 8M floats = 32MB
  float* q    = (float*)d_ws;                        // [B,L,HID]
  float* k    = q + nact;                            // [B,L,HID]
  float* v    = k + nact;                            // [B,L,HID] -> becomes y
  float* beta = v + nact;                            // [B,H,L]
  // ws needed: 3*32MB + 0.25MB ~= 96.3 MB

  const int M = B_ * L_, N = HID_, K = HID_;
  dim3 gg(N / TN, M / TM);                           // (32, 32)

  gemm_f32_wmma<<<gg, 256, 0, stream>>>(x, Wq, nullptr, q, M, N, K);
  gemm_f32_wmma<<<gg, 256, 0, stream>>>(x, Wk, nullptr, k, M, N, K);
  gemm_f32_wmma<<<gg, 256, 0, stream>>>(x, Wv, nullptr, v, M, N, K);

  beta_kernel<<<B_ * L_, 256, 0, stream>>>(x, Wg, bg, beta);
  rope_phi_kernel<<<B_ * L_ * H_, 64, 0, stream>>>(q, k);

  scan_kernel<<<B_ * H_, D_, 0, stream>>>(q, k, v, beta);

  gemm_f32_wmma<<<gg, 256, 0, stream>>>(v, Wo, bo, out, M, N, K);
}